// KoopmanOperator_36309653520440
// MI455X (gfx1250) — compile-verified
//
#include <hip/hip_runtime.h>

// CDNA5 / gfx1250 Koopman operator rollout: y_{t+1} = y_t * W^T, store every step.
// fp32 WMMA (16x16x4) — kernel is HBM-write bound (1.07 GB out, ~46us floor at
// 23.3 TB/s), so full fp32 precision is free. W^T fragments live in VGPRs for all
// 512 steps; the 16x256 state ping-pongs through padded LDS buffers.
//
// Round-2 change: replace __syncthreads() (which waits on ALL memory counters,
// serializing the non-temporal HBM stores into every one of the 512 dependent
// steps) with an LDS-only sync: s_wait_dscnt 0 + split workgroup barrier
// (s_barrier_signal/-wait -1). Global NT stores are write-once to disjoint
// addresses and need no ordering — they stay in flight across steps, keeping
// the write pipeline at the bandwidth roofline.

typedef float v2f __attribute__((ext_vector_type(2)));
typedef float v8f __attribute__((ext_vector_type(8)));

#define KD 256
#define TS 512
#define PITCH 260   // padded LDS row pitch (floats): conflict-free ds_load_b64 / stores

// LDS-only workgroup barrier: order DS ops, let VMEM stores drift.
__device__ __forceinline__ void lds_barrier() {
    asm volatile(
        "s_wait_dscnt 0x0\n\t"
        "s_barrier_signal -1\n\t"
        "s_barrier_wait -1"
        ::: "memory");
}

__global__ __launch_bounds__(256, 1)
void koopman_rollout_kernel(const float* __restrict__ x,
                            const float* __restrict__ W,
                            float* __restrict__ out) {
    __shared__ float lds[2][16 * PITCH];

    const int tid   = threadIdx.x;
    const int wave  = tid >> 5;        // 0..7  -> owns N columns [32*wave, 32*wave+32)
    const int lane  = tid & 31;
    const int lhalf = lane >> 4;       // 0 or 1 (lane half selects K pair / M half)
    const int l16   = lane & 15;

    const int rowBase = blockIdx.x * 16;   // this WG's 16 batch rows

    // ---- stage y0 (16 x 256) into lds[0], coalesced float4 loads ----
    {
        const float4* src = (const float4*)(x + (size_t)rowBase * KD);
        #pragma unroll
        for (int i = 0; i < 4; ++i) {
            int e = i * 256 + tid;            // float4 index in 16x256 tile
            float4 v = src[e];
            int m = e >> 6;                   // row
            int k = (e & 63) << 2;            // col
            float* dst = &lds[0][m * PITCH + k];
            dst[0] = v.x; dst[1] = v.y; dst[2] = v.z; dst[3] = v.w;
        }
    }

    // ---- preload B = W^T slice for this wave into registers (persists all T steps)
    // B fragment layout for v_wmma_f32_16x16x4_f32:
    //   lanes 0-15 : V0=B[k0][n],   V1=B[k0+1][n]   (n = l16)
    //   lanes 16-31: V0=B[k0+2][n], V1=B[k0+3][n]
    // B[k][n] = W[ncol][k]  -> contiguous float2 from row-major W.
    v2f Breg[2][64];
    #pragma unroll
    for (int nt = 0; nt < 2; ++nt) {
        const int ncol = wave * 32 + nt * 16 + l16;
        const float* wrow = W + (size_t)ncol * KD;
        #pragma unroll
        for (int c = 0; c < 64; ++c) {
            const int k = c * 4 + lhalf * 2;
            Breg[nt][c] = *(const v2f*)(wrow + k);
        }
    }

    lds_barrier();   // y0 staged (ds stores drained) before anyone reads it

    const int nc0 = wave * 32 + l16;   // N column of this lane, tile 0
    const int nc1 = nc0 + 16;          // tile 1
    const size_t obase0 = (size_t)rowBase * (size_t)(TS * KD);

    int cur = 0;
    #pragma unroll 1
    for (int t = 0; t < TS; ++t) {
        v8f acc0 = {};
        v8f acc1 = {};

        // A fragment base: lane L reads row (L%16), K offset 2*(L/16); pitch 260
        // makes all 32 lanes hit distinct banks (b64 loads, fully conflict-free).
        const float* abuf = &lds[cur][l16 * PITCH + lhalf * 2];

        #pragma unroll
        for (int c = 0; c < 64; ++c) {
            v2f a = *(const v2f*)(abuf + c * 4);
            acc0 = __builtin_amdgcn_wmma_f32_16x16x4_f32(
                       false, a, false, Breg[0][c], (short)0, acc0, false, false);
            acc1 = __builtin_amdgcn_wmma_f32_16x16x4_f32(
                       false, a, false, Breg[1][c], (short)0, acc1, false, false);
        }

        // D layout: VGPR r, lanes 0-15 -> row r, lanes 16-31 -> row r+8; col = l16.
        float* nbuf = lds[cur ^ 1];
        const size_t obt = obase0 + (size_t)t * KD;
        #pragma unroll
        for (int r = 0; r < 8; ++r) {
            const int m = r + 8 * lhalf;
            nbuf[m * PITCH + nc0] = acc0[r];
            nbuf[m * PITCH + nc1] = acc1[r];
            float* orow = out + obt + (size_t)m * (size_t)(TS * KD);
            __builtin_nontemporal_store(acc0[r], orow + nc0);  // write-once output:
            __builtin_nontemporal_store(acc1[r], orow + nc1);  // keep out of caches,
        }                                                      // no sync needed

        // Order only LDS across the step boundary; NT stores stay in flight.
        lds_barrier();
        cur ^= 1;
    }
}

extern "C" void kernel_launch(void* const* d_in, const int* in_sizes, int n_in,
                              void* d_out, int out_size, void* d_ws, size_t ws_size,
                              hipStream_t stream) {
    const float* x = (const float*)d_in[0];   // (2048, 1, 256) f32
    const float* W = (const float*)d_in[1];   // (256, 256) f32
    // d_in[2] = T (scalar 512) — baked in as TS.
    float* out = (float*)d_out;               // (2048, 512, 256) f32

    dim3 grid(2048 / 16);   // 128 workgroups, 16 batch rows each
    dim3 block(256);        // 8 waves: each owns a 32-column N slice
    koopman_rollout_kernel<<<grid, block, 0, stream>>>(x, W, out);
}